// ReactionMapper_14353780703958
// MI455X (gfx1250) — compile-verified
//
#include <hip/hip_runtime.h>
#include <hip/hip_bf16.h>
#include <math.h>

typedef __attribute__((ext_vector_type(2))) float v2f;
typedef __attribute__((ext_vector_type(8))) float v8f;

#define D_FEAT 128
#define D_HID  64

// ---------------------------------------------------------------------------
// Kernel 0: zero output, init segment-max to -inf and segment-denom to 0.
// ---------------------------------------------------------------------------
__global__ void rm_init(float* __restrict__ out, int out_n,
                        float* __restrict__ gmax, float* __restrict__ denom, int nrxn) {
    int i = blockIdx.x * blockDim.x + threadIdx.x;
    if (i < out_n) out[i] = 0.0f;
    if (i < nrxn) {
        gmax[i]  = -INFINITY;
        denom[i] = 0.0f;
    }
}

// ---------------------------------------------------------------------------
// Kernel 1: per-metabolite transform + gate using V_WMMA_F32_16X16X4_F32.
// One wave (32 lanes) computes a 16-row block:
//   T_met[16,128] = relu(MF[16,128] @ Wt[128,128] + bt)
//   gate_met[16]  = relu(MF[16,128] @ W1[128,64] + b1) @ W2[64] + b2
// A fragment layout (16x4 f32): lane = half*16 + m; a[v] = A[m][k0 + 2*half + v]
// B fragment layout (4x16 f32): b[v] = B[k0 + 2*half + v][n], n = lane & 15
// C/D layout (16x16 f32): c[r] = C[r + 8*half][lane & 15]
// ---------------------------------------------------------------------------
__global__ __launch_bounds__(32) void rm_transform_gate(
    const float* __restrict__ MF,
    const float* __restrict__ W1, const float* __restrict__ b1,
    const float* __restrict__ W2, const float* __restrict__ b2,
    const float* __restrict__ Wt, const float* __restrict__ bt,
    float* __restrict__ Tm, float* __restrict__ gatem, int nmet) {

    const int m0 = blockIdx.x * 16;
    if (m0 >= nmet) return;                 // uniform per wave -> EXEC all ones inside

    const int lane = threadIdx.x & 31;
    const int half = lane >> 4;             // 0: K even-pair / rows 0-7, 1: K odd-pair / rows 8-15
    const int mr   = lane & 15;             // m for A, n for B/C/D

    // ---- preload the full 16x128 A block: 32 k-steps x v2f = 64 VGPRs/lane ----
    v2f a[32];
    const float* arow = MF + (size_t)(m0 + mr) * D_FEAT + 2 * half;
    #pragma unroll
    for (int ks = 0; ks < 32; ++ks) {
        a[ks].x = arow[4 * ks + 0];
        a[ks].y = arow[4 * ks + 1];
    }

    // ---- transform_nn: 8 n-tiles over Wt[128,128] ----
    for (int nt = 0; nt < 8; ++nt) {
        v8f acc = {};
        const float* bp = Wt + (size_t)(2 * half) * D_FEAT + nt * 16 + mr;
        #pragma unroll
        for (int ks = 0; ks < 32; ++ks) {
            v2f b;
            b.x = bp[(size_t)(4 * ks + 0) * D_FEAT];
            b.y = bp[(size_t)(4 * ks + 1) * D_FEAT];
            acc = __builtin_amdgcn_wmma_f32_16x16x4_f32(
                false, a[ks], false, b, (short)0, acc, false, false);
        }
        const int n = nt * 16 + mr;
        const float bias = bt[n];
        #pragma unroll
        for (int r = 0; r < 8; ++r) {
            float t = acc[r] + bias;
            t = t > 0.0f ? t : 0.0f;
            Tm[(size_t)(m0 + r + 8 * half) * D_FEAT + n] = t;
        }
    }

    // ---- gate_nn: 4 n-tiles over W1[128,64], then dot with W2[64] ----
    float g[8];
    #pragma unroll
    for (int r = 0; r < 8; ++r) g[r] = 0.0f;

    for (int nt = 0; nt < 4; ++nt) {
        v8f acc = {};
        const float* bp = W1 + (size_t)(2 * half) * D_HID + nt * 16 + mr;
        #pragma unroll
        for (int ks = 0; ks < 32; ++ks) {
            v2f b;
            b.x = bp[(size_t)(4 * ks + 0) * D_HID];
            b.y = bp[(size_t)(4 * ks + 1) * D_HID];
            acc = __builtin_amdgcn_wmma_f32_16x16x4_f32(
                false, a[ks], false, b, (short)0, acc, false, false);
        }
        const int n = nt * 16 + mr;
        const float bb = b1[n];
        const float w2 = W2[n];
        #pragma unroll
        for (int r = 0; r < 8; ++r) {
            float h = acc[r] + bb;
            h = h > 0.0f ? h : 0.0f;
            g[r] += h * w2;
        }
    }

    // reduce gate partials across the 16 lanes of each half
    const float bias2 = b2[0];
    #pragma unroll
    for (int r = 0; r < 8; ++r) {
        float v = g[r];
        v += __shfl_xor(v, 1, 16);
        v += __shfl_xor(v, 2, 16);
        v += __shfl_xor(v, 4, 16);
        v += __shfl_xor(v, 8, 16);
        if (mr == 0) gatem[m0 + r + 8 * half] = v + bias2;
    }
}

// ---------------------------------------------------------------------------
// Sign-aware float atomic max via integer atomics (monotone bit ordering).
// ---------------------------------------------------------------------------
__device__ __forceinline__ void atomicMaxFloat(float* addr, float v) {
    if (v >= 0.0f) atomicMax((int*)addr, __float_as_int(v));
    else           atomicMin((unsigned int*)addr, __float_as_uint(v));
}

// Kernel 2: segment max of gate over reactions.
__global__ void rm_gate_max(const int* __restrict__ he, int E,
                            const float* __restrict__ gatem, float* __restrict__ gmax) {
    int e = blockIdx.x * blockDim.x + threadIdx.x;
    if (e >= E) return;
    int m = he[e];
    int r = he[E + e];
    atomicMaxFloat(&gmax[r], gatem[m]);
}

// Kernel 3: segment sum of exp(gate - gmax).
__global__ void rm_gate_denom(const int* __restrict__ he, int E,
                              const float* __restrict__ gatem,
                              const float* __restrict__ gmax, float* __restrict__ denom) {
    int e = blockIdx.x * blockDim.x + threadIdx.x;
    if (e >= E) return;
    int m = he[e];
    int r = he[E + e];
    atomicAdd(&denom[r], __expf(gatem[m] - gmax[r]));
}

// ---------------------------------------------------------------------------
// Kernel 4: attention-weighted scatter. One wave per edge; each lane handles
// 4 columns (float4 coalesced gather from L2-resident T_met, 4 atomic f32
// adds into L2-resident Z_r).
// ---------------------------------------------------------------------------
__global__ __launch_bounds__(256) void rm_scatter(
    const int* __restrict__ he, int E,
    const float* __restrict__ gatem, const float* __restrict__ gmax,
    const float* __restrict__ denom, const float* __restrict__ Tm,
    float* __restrict__ out) {

    const int wave = blockIdx.x * (blockDim.x >> 5) + (threadIdx.x >> 5);
    const int lane = threadIdx.x & 31;
    if (wave >= E) return;                  // uniform per wave

    const int m = he[wave];
    const int r = he[E + wave];
    const float alpha = __expf(gatem[m] - gmax[r]) / denom[r];

    const float4 t = *(const float4*)(Tm + (size_t)m * D_FEAT + lane * 4);
    float* op = out + (size_t)r * D_FEAT + lane * 4;
    atomicAdd(op + 0, alpha * t.x);
    atomicAdd(op + 1, alpha * t.y);
    atomicAdd(op + 2, alpha * t.z);
    atomicAdd(op + 3, alpha * t.w);
}

// ---------------------------------------------------------------------------
extern "C" void kernel_launch(void* const* d_in, const int* in_sizes, int n_in,
                              void* d_out, int out_size, void* d_ws, size_t ws_size,
                              hipStream_t stream) {
    const float* MF = (const float*)d_in[0];
    const int*   he = (const int*)d_in[1];     // hyperedge_index [2, E]
    const float* W1 = (const float*)d_in[2];
    const float* b1 = (const float*)d_in[3];
    const float* W2 = (const float*)d_in[4];
    const float* b2 = (const float*)d_in[5];
    const float* Wt = (const float*)d_in[6];
    const float* bt = (const float*)d_in[7];
    float* out = (float*)d_out;

    const int nmet = in_sizes[0] / D_FEAT;
    const int E    = in_sizes[1] / 2;
    const int nrxn = out_size / D_FEAT;

    // workspace layout (16B aligned)
    char* ws = (char*)d_ws;
    size_t off = 0;
    float* Tm    = (float*)(ws + off); off += (((size_t)nmet * D_FEAT * 4) + 15) & ~(size_t)15;
    float* gatem = (float*)(ws + off); off += (((size_t)nmet * 4) + 15) & ~(size_t)15;
    float* gmax  = (float*)(ws + off); off += (((size_t)nrxn * 4) + 15) & ~(size_t)15;
    float* denom = (float*)(ws + off); off += (((size_t)nrxn * 4) + 15) & ~(size_t)15;

    // 0) init output / segment state
    {
        int n = out_size > nrxn ? out_size : nrxn;
        rm_init<<<(n + 255) / 256, 256, 0, stream>>>(out, out_size, gmax, denom, nrxn);
    }
    // 1) per-metabolite transform + gate (WMMA fp32)
    {
        int blocks = (nmet + 15) / 16;
        rm_transform_gate<<<blocks, 32, 0, stream>>>(MF, W1, b1, W2, b2, Wt, bt,
                                                     Tm, gatem, nmet);
    }
    // 2) segment max
    rm_gate_max<<<(E + 255) / 256, 256, 0, stream>>>(he, E, gatem, gmax);
    // 3) segment exp-sum
    rm_gate_denom<<<(E + 255) / 256, 256, 0, stream>>>(he, E, gatem, gmax, denom);
    // 4) weighted scatter (one wave per edge)
    rm_scatter<<<(E + 7) / 8, 256, 0, stream>>>(he, E, gatem, gmax, denom, Tm, out);
}